// SWRecursiveNN_23733989278140
// MI455X (gfx1250) — compile-verified
//
#include <hip/hip_runtime.h>
#include <hip/hip_bf16.h>

// ---------------------------------------------------------------------------
// RNTN over a complete binary tree, CDNA5 (gfx1250) WMMA implementation.
//   states[i]     : [32,32] fp32 per node, in d_ws
//   leaves        : states[i] = relu(emb[words[i]])
//   internal      : states[i] = relu( l @ (Wn @ r) + bias ), Wn = emb[words[i]]
//   output        : prediction (as float), loss
// One wave32 per internal node; each 32x32x32 fp32 matmul = 4 C-tiles x 8
// v_wmma_f32_16x16x4_f32 steps. Intermediate T = Wn @ r staged through LDS.
// ---------------------------------------------------------------------------

typedef __attribute__((ext_vector_type(2))) float v2f;
typedef __attribute__((ext_vector_type(8))) float v8f;

#define EDIM 32

// ---------------------------------------------------------------------------
// Leaves: states[leaf] = relu(emb[words[leaf]])
// ---------------------------------------------------------------------------
__global__ __launch_bounds__(256)
void rntn_leaf_kernel(float* __restrict__ states,
                      const float* __restrict__ emb,
                      const int* __restrict__ words)
{
    const int leaf = blockIdx.x;
    const int t    = threadIdx.x;                 // 0..255, one float4 each
    const float4* src = (const float4*)(emb + (size_t)words[leaf] * (EDIM * EDIM));
    float4*       dst = (float4*)(states + (size_t)leaf * (EDIM * EDIM));
    float4 v = src[t];
    v.x = fmaxf(v.x, 0.0f);
    v.y = fmaxf(v.y, 0.0f);
    v.z = fmaxf(v.z, 0.0f);
    v.w = fmaxf(v.w, 0.0f);
    dst[t] = v;
}

// ---------------------------------------------------------------------------
// One tree level. Block = 1 wave32 = 1 internal node.
//   T    = Wn @ r                (32x32x32 fp32, 32 WMMA ops, staged to LDS)
//   comp = l @ T + bias ; relu   (another 32 WMMA ops)
// WMMA fp32 operand layouts (ISA 7.12.2):
//   A 16x4 : lane half h, vgpr v -> K = 2h + v, M = lane&15
//   B 4x16 : lane half h, vgpr v -> K = 2h + v, N = lane&15
//   C 16x16: vgpr v, half h      -> M = v + 8h, N = lane&15
// ---------------------------------------------------------------------------
__global__ __launch_bounds__(32)
void rntn_level_kernel(float* __restrict__ states,
                       const float* __restrict__ emb,
                       const int* __restrict__ words,
                       const int* __restrict__ left,
                       const int* __restrict__ right,
                       const float* __restrict__ bias,
                       int base)
{
    const int node = base + blockIdx.x;
    const int lane = threadIdx.x;        // 0..31
    const int half = lane >> 4;          // 0 or 1
    const int ln   = lane & 15;          // 0..15

    const float* Wn = emb    + (size_t)words[node] * (EDIM * EDIM);
    const float* lm = states + (size_t)left[node]  * (EDIM * EDIM);
    const float* rm = states + (size_t)right[node] * (EDIM * EDIM);
    float*       dm = states + (size_t)node        * (EDIM * EDIM);

    __shared__ float T[EDIM][EDIM + 1];  // padded to soften bank conflicts

    // ---------------- T = Wn @ r ----------------
    #pragma unroll
    for (int ti = 0; ti < 2; ++ti) {
        #pragma unroll
        for (int tj = 0; tj < 2; ++tj) {
            v8f acc = {};
            #pragma unroll
            for (int kb = 0; kb < 8; ++kb) {
                const int k0 = 4 * kb + 2 * half;       // K of vgpr 0
                v2f a;
                a.x = Wn[(16 * ti + ln) * EDIM + k0 + 0];
                a.y = Wn[(16 * ti + ln) * EDIM + k0 + 1];
                v2f b;
                b.x = rm[(k0 + 0) * EDIM + 16 * tj + ln];
                b.y = rm[(k0 + 1) * EDIM + 16 * tj + ln];
                acc = __builtin_amdgcn_wmma_f32_16x16x4_f32(
                        /*neg_a=*/false, a, /*neg_b=*/false, b,
                        /*c_mod=*/(short)0, acc,
                        /*reuse_a=*/false, /*reuse_b=*/false);
            }
            #pragma unroll
            for (int v = 0; v < 8; ++v)
                T[16 * ti + 8 * half + v][16 * tj + ln] = acc[v];
        }
    }
    __syncthreads();

    // ---------------- comp = l @ T + bias ; relu ----------------
    #pragma unroll
    for (int ti = 0; ti < 2; ++ti) {
        #pragma unroll
        for (int tj = 0; tj < 2; ++tj) {
            v8f acc;
            #pragma unroll
            for (int v = 0; v < 8; ++v)
                acc[v] = bias[(16 * ti + 8 * half + v) * EDIM + 16 * tj + ln];
            #pragma unroll
            for (int kb = 0; kb < 8; ++kb) {
                const int k0 = 4 * kb + 2 * half;
                v2f a;
                a.x = lm[(16 * ti + ln) * EDIM + k0 + 0];
                a.y = lm[(16 * ti + ln) * EDIM + k0 + 1];
                v2f b;
                b.x = T[k0 + 0][16 * tj + ln];
                b.y = T[k0 + 1][16 * tj + ln];
                acc = __builtin_amdgcn_wmma_f32_16x16x4_f32(
                        false, a, false, b, (short)0, acc, false, false);
            }
            #pragma unroll
            for (int v = 0; v < 8; ++v) {
                float x = acc[v];
                dm[(16 * ti + 8 * half + v) * EDIM + 16 * tj + ln] =
                    x > 0.0f ? x : 0.0f;
            }
        }
    }
}

// ---------------------------------------------------------------------------
// Projection + cross-entropy + argmax. One block of 128 threads (LBL=128).
//   logits[j] = root . W_proj[j] + b_proj[j]
//   out[0] = argmax(logits) (as float); out[1] = -(log_softmax(logits)[label])
// ---------------------------------------------------------------------------
__global__ __launch_bounds__(128)
void rntn_project_kernel(const float* __restrict__ root,     // [1024]
                         const float* __restrict__ W_proj,   // [128,1024]
                         const float* __restrict__ b_proj,   // [128]
                         const int*   __restrict__ label_p,  // [1]
                         float* __restrict__ out,            // [2]
                         int lbl_count)
{
    const int j = threadIdx.x;
    __shared__ float logits[128];

    float s = b_proj[j];
    const float* wrow = W_proj + (size_t)j * (EDIM * EDIM);
    #pragma unroll 8
    for (int k = 0; k < EDIM * EDIM; ++k)
        s = fmaf(root[k], wrow[k], s);
    logits[j] = s;
    __syncthreads();

    if (j == 0) {
        float mx = logits[0];
        int   arg = 0;
        for (int i = 1; i < lbl_count; ++i) {
            if (logits[i] > mx) { mx = logits[i]; arg = i; }
        }
        float sum = 0.0f;
        for (int i = 0; i < lbl_count; ++i)
            sum += __expf(logits[i] - mx);
        const int label = label_p[0];
        const float logsm = logits[label] - mx - __logf(sum);
        out[0] = (float)arg;
        out[1] = -logsm;
    }
}

// ---------------------------------------------------------------------------
// kernel_launch
//   d_in: 0 words[N], 1 left[N], 2 right[N], 3 is_leaf[N] (unused),
//         4 emb[V*1024], 5 bias[1024], 6 W_proj[128*1024], 7 b_proj[128],
//         8 label[1]
//   d_ws: states buffer, N * 1024 floats (~16.8 MB)
// ---------------------------------------------------------------------------
extern "C" void kernel_launch(void* const* d_in, const int* in_sizes, int n_in,
                              void* d_out, int out_size, void* d_ws, size_t ws_size,
                              hipStream_t stream)
{
    const int*   words  = (const int*)d_in[0];
    const int*   left   = (const int*)d_in[1];
    const int*   right  = (const int*)d_in[2];
    const float* emb    = (const float*)d_in[4];
    const float* bias   = (const float*)d_in[5];
    const float* W_proj = (const float*)d_in[6];
    const float* b_proj = (const float*)d_in[7];
    const int*   label  = (const int*)d_in[8];

    const int N  = in_sizes[0];        // 4095
    const int Lf = (N + 1) / 2;        // 2048 leaves
    const int lbl_count = in_sizes[7]; // 128

    float* states = (float*)d_ws;

    // Leaves: relu(embedding gather)
    rntn_leaf_kernel<<<Lf, 256, 0, stream>>>(states, emb, words);

    // Tree levels: 1024, 512, ..., 1 nodes; one wave32 per node.
    int base  = Lf;
    int count = Lf / 2;
    while (count >= 1) {
        rntn_level_kernel<<<count, 32, 0, stream>>>(
            states, emb, words, left, right, bias, base);
        base  += count;
        count >>= 1;
    }

    // Projection + loss from the root node (last node).
    rntn_project_kernel<<<1, 128, 0, stream>>>(
        states + (size_t)(N - 1) * (EDIM * EDIM),
        W_proj, b_proj, label, (float*)d_out, lbl_count);
}